// Kmeans_9062380995191
// MI455X (gfx1250) — compile-verified
//
#include <hip/hip_runtime.h>

typedef __attribute__((ext_vector_type(2))) float v2f;
typedef __attribute__((ext_vector_type(4))) float v4f;
typedef __attribute__((ext_vector_type(8))) float v8f;

constexpr int Bq = 4, Hq = 8, Lq = 4096, Dq = 64, Cq = 512;
constexpr int ROWS_PER_BLOCK = 128;
constexpr int LDS_STRIDE = 68;            // floats; pad 64->68 kills LDS bank conflicts
constexpr float COMMIT = 0.0001f;
constexpr float EPSQ = 1e-12f;
constexpr long long LOSS_IDX = (long long)Bq * Hq * Lq * Cq;

__global__ void init_loss_kernel(float* out) {
    if (threadIdx.x == 0) out[LOSS_IDX] = 0.0f;
}

// msq[h*C + c] = ||means[h,c,:]||^2   (4096 entries into d_ws)
__global__ void msq_kernel(const float* __restrict__ means, float* __restrict__ msq) {
    int i = blockIdx.x * blockDim.x + threadIdx.x;   // 0..Hq*Cq-1
    const v4f* p = (const v4f*)(means + (size_t)i * Dq);
    float s = 0.0f;
#pragma unroll
    for (int j = 0; j < Dq / 4; ++j) {
        v4f v = p[j];
        s += v.x * v.x + v.y * v.y + v.z * v.z + v.w * v.w;
    }
    msq[i] = s;
}

__launch_bounds__(256, 2)
__global__ void kmeans_wmma_kernel(const float* __restrict__ x,
                                   const float* __restrict__ means,
                                   const float* __restrict__ msq,
                                   float* __restrict__ out) {
    __shared__ float Axn[ROWS_PER_BLOCK * LDS_STRIDE];
    __shared__ float xnsq[ROWS_PER_BLOCK];

    const int t    = threadIdx.x;
    const int lblk = blockIdx.x;           // l-tile: 0..Lq/128-1
    const int bh   = blockIdx.y;           // fused (b,h): 0..31
    const int h    = bh % Hq;
    const long long l0 = (long long)lblk * ROWS_PER_BLOCK;

    // ---- Stage 1: coalesced NT load of x tile (128x64 f32) into padded LDS ----
    const v4f* xbase4 = (const v4f*)(x + ((long long)bh * Lq + l0) * Dq);
#pragma unroll
    for (int i = 0; i < 8; ++i) {
        int f   = i * 256 + t;             // float4 index in 128x64 tile
        int row = f >> 4;                  // 16 float4 per row
        int col = (f & 15) << 2;
        v4f v = __builtin_nontemporal_load(xbase4 + f);
        *(v4f*)&Axn[row * LDS_STRIDE + col] = v;
    }
    __syncthreads();

    // ---- Stage 2: per-row L2 normalization in LDS ----
    if (t < ROWS_PER_BLOCK) {
        float* rp = &Axn[t * LDS_STRIDE];
        float s = 0.0f;
#pragma unroll
        for (int j = 0; j < Dq; ++j) s += rp[j] * rp[j];
        float inv = 1.0f / fmaxf(sqrtf(s), EPSQ);
#pragma unroll
        for (int j = 0; j < Dq; ++j) rp[j] *= inv;
        xnsq[t] = s * inv * inv;           // ||xn||^2 (==1 unless degenerate)
    }
    __syncthreads();

    // ---- Stage 3: WMMA GEMM, each wave owns a 16-row strip over all 512 cols ----
    const int wave = t >> 5;
    const int lane = t & 31;
    const int rl   = lane & 15;            // row-in-tile (A) / col-in-tile (B,D)
    const int half = lane >> 4;
    const int k0   = half * 2;             // f32 16x16x4: lanes16-31 carry K=2,3
    const int wrow = wave * 16;

    // A fragments for all 16 K-steps, loaded once from LDS (conflict-free)
    v2f afrag[16];
#pragma unroll
    for (int ks = 0; ks < 16; ++ks) {
        const float* ap = &Axn[(wrow + rl) * LDS_STRIDE + ks * 4 + k0];
        afrag[ks] = v2f{ap[0], ap[1]};
    }

    const float* mh = means + (size_t)h * Cq * Dq;
    float* outbase = out + ((long long)bh * Lq + l0 + wrow) * Cq;

    float maxv[8];
    int   maxi[8];
#pragma unroll
    for (int i = 0; i < 8; ++i) { maxv[i] = -3.4e38f; maxi[i] = 0; }

    for (int nt = 0; nt < Cq / 16; ++nt) {
        const int cbase = nt * 16;
        const float* bp = mh + (size_t)(cbase + rl) * Dq + k0;  // means row = B column

        v8f acc = {0.f, 0.f, 0.f, 0.f, 0.f, 0.f, 0.f, 0.f};
#pragma unroll
        for (int ks = 0; ks < 16; ++ks) {
            v2f bfrag = v2f{bp[ks * 4], bp[ks * 4 + 1]};
            acc = __builtin_amdgcn_wmma_f32_16x16x4_f32(
                false, afrag[ks], false, bfrag, (short)0, acc, false, false);
        }

        // D layout: VGPR i -> row i + 8*half, col rl. NT store + argmax tracking.
#pragma unroll
        for (int i = 0; i < 8; ++i) {
            int rg = i + half * 8;
            float v = acc[i];
            __builtin_nontemporal_store(v, outbase + (long long)rg * Cq + cbase + rl);
            if (v > maxv[i]) { maxv[i] = v; maxi[i] = cbase + rl; }
        }
    }

    // ---- Stage 4: argmax reduce across the 16 lanes of each half-wave ----
#pragma unroll
    for (int m = 1; m < 16; m <<= 1) {
#pragma unroll
        for (int i = 0; i < 8; ++i) {
            float ov = __shfl_xor(maxv[i], m, 32);
            int   oi = __shfl_xor(maxi[i], m, 32);
            if (ov > maxv[i] || (ov == maxv[i] && oi < maxi[i])) {
                maxv[i] = ov; maxi[i] = oi;
            }
        }
    }

    // ---- Stage 5: loss contribution: ||xn||^2 - 2*max + ||means[argmax]||^2 ----
    if (rl == 0) {
        const float* msqh = msq + (size_t)h * Cq;
        float contrib = 0.0f;
#pragma unroll
        for (int i = 0; i < 8; ++i) {
            int r = wrow + i + half * 8;
            contrib += xnsq[r] - 2.0f * maxv[i] + msqh[maxi[i]];
        }
        contrib *= COMMIT / ((float)Bq * (float)Hq * (float)Lq * (float)Dq);
        atomicAdd(out + LOSS_IDX, contrib);
    }
}

extern "C" void kernel_launch(void* const* d_in, const int* in_sizes, int n_in,
                              void* d_out, int out_size, void* d_ws, size_t ws_size,
                              hipStream_t stream) {
    const float* x     = (const float*)d_in[0];
    const float* means = (const float*)d_in[1];
    float* out = (float*)d_out;
    float* msq = (float*)d_ws;            // 4096 floats = 16 KB scratch

    init_loss_kernel<<<1, 32, 0, stream>>>(out);
    msq_kernel<<<(Hq * Cq) / 256, 256, 0, stream>>>(means, msq);

    dim3 grid(Lq / ROWS_PER_BLOCK, Bq * Hq);
    kmeans_wmma_kernel<<<grid, 256, 0, stream>>>(x, means, msq, out);
}